// GAT_80006650790247
// MI455X (gfx1250) — compile-verified
//
#include <hip/hip_runtime.h>
#include <hip/hip_bf16.h>
#include <math.h>

typedef __attribute__((ext_vector_type(2))) float v2f;
typedef __attribute__((ext_vector_type(8))) float v8f;

#define NEG_SLOPE 0.2f

// ---------- order-preserving float<->uint encoding for atomicMax ----------
__device__ __forceinline__ unsigned enc_f32(float f) {
    unsigned u = __float_as_uint(f);
    return (u & 0x80000000u) ? ~u : (u | 0x80000000u);
}
__device__ __forceinline__ float dec_f32(unsigned u) {
    unsigned v = (u & 0x80000000u) ? (u & 0x7fffffffu) : ~u;
    return __uint_as_float(v);
}

// ---------- pack/pad B: Bpack[kk][c][01] = { B[2kk][c], B[2kk+1][c] }, c >= Dout -> 0 ----------
// Gives each WMMA lane its two K-values as one aligned 8-byte load.
__global__ void gat_pack_b(const float* __restrict__ B, float* __restrict__ Bp,
                           int K, int Dout, int DP) {
    const int i = blockIdx.x * blockDim.x + threadIdx.x;   // over (K/2)*DP
    if (i >= (K / 2) * DP) return;
    const int kk = i / DP;
    const int c  = i - kk * DP;
    float x0 = 0.f, x1 = 0.f;
    if (c < Dout) {
        x0 = B[(size_t)(2 * kk) * Dout + c];
        x1 = B[(size_t)(2 * kk + 1) * Dout + c];
    }
    Bp[2 * i]     = x0;
    Bp[2 * i + 1] = x1;
}

// ---------- WMMA GEMM: C[N,DP] = A[N,K] * B[K,DP] (B pre-packed/padded) ----------
// fp32 v_wmma_f32_16x16x4_f32. Block = 8 waves; B staged in LDS once per block;
// each wave computes a full 16-row x (NT*16)-col strip (NT accumulators), so one
// A b64 load feeds NT WMMAs. No per-lane guards anywhere: row tile start is
// uniformly clamped, B is zero-padded. EXEC is all-ones for every WMMA.
template<int K, int NT>
__global__ __launch_bounds__(256) void gat_gemm_wmma(const float* __restrict__ A,
                                                     const float* __restrict__ Bp,
                                                     float* __restrict__ C, int N) {
    constexpr int DP = NT * 16;
    __shared__ float Bs[K * DP];           // packed-pair layout, same as Bp

    // cooperative stage of B into LDS (K*DP floats, 256 threads, float4 chunks)
    const int tid = threadIdx.y * 32 + threadIdx.x;
    #pragma unroll
    for (int i = tid * 4; i < K * DP; i += 256 * 4)
        *(float4*)&Bs[i] = *(const float4*)&Bp[i];
    __syncthreads();

    const int Mtiles = (N + 15) >> 4;
    const int strip = blockIdx.x * blockDim.y + threadIdx.y;   // uniform per wave
    if (strip >= Mtiles) return;
    int row0 = strip << 4;
    if (row0 + 16 > N) row0 = N - 16;      // uniform clamp: full loads stay in range

    const int m  = threadIdx.x & 15;       // A row / B,C col within tile
    const int kh = threadIdx.x >> 4;       // K half for A/B, M half for C/D
    const float* __restrict__ arow = A + (size_t)(row0 + m) * K;

    v8f acc[NT];
    #pragma unroll
    for (int t = 0; t < NT; ++t) acc[t] = (v8f){0.f,0.f,0.f,0.f,0.f,0.f,0.f,0.f};

    #pragma unroll
    for (int k = 0; k < K; k += 4) {
        const int ka = k + 2 * kh;
        const v2f a = *(const v2f*)(arow + ka);          // global_load_b64
        const int kk = ka >> 1;
        #pragma unroll
        for (int t = 0; t < NT; ++t) {
            const v2f b = *(const v2f*)&Bs[(kk * DP + t * 16 + m) * 2];  // ds_load_b64
            acc[t] = __builtin_amdgcn_wmma_f32_16x16x4_f32(
                false, a, false, b, (short)0, acc[t], false, false);
        }
    }

    // store: ISA C/D layout -> VGPR v holds row (v + 8*kh), col = m per tile
    #pragma unroll
    for (int t = 0; t < NT; ++t) {
        const int cn = t * 16 + m;
        #pragma unroll
        for (int v = 0; v < 8; ++v)
            C[(size_t)(row0 + v + 8 * kh) * DP + cn] = acc[t][v];
    }
}

// ---------- per-node attention scores: el[i]=feat[i,:].al, er[i]=feat[i,:].ar ----------
__global__ void gat_attn_scores(const float* __restrict__ F,
                                const float* __restrict__ al,
                                const float* __restrict__ ar,
                                float* __restrict__ el, float* __restrict__ er,
                                int N, int D, int FS) {
    const int i = blockIdx.x * blockDim.x + threadIdx.x;
    if (i >= N) return;
    const float* f = F + (size_t)i * FS;
    float a = 0.f, b = 0.f;
    for (int j = 0; j < D; j += 4) {
        float4 v = *(const float4*)(f + j);
        a += v.x * al[j] + v.y * al[j + 1] + v.z * al[j + 2] + v.w * al[j + 3];
        b += v.x * ar[j] + v.y * ar[j + 1] + v.z * ar[j + 2] + v.w * ar[j + 3];
    }
    el[i] = a;
    er[i] = b;
}

// ---------- edge pass 1: e = leaky_relu(el[src]+er[dst]); segment max over dst ----------
__global__ void gat_edge_max(const int* __restrict__ src, const int* __restrict__ dst,
                             const float* __restrict__ el, const float* __restrict__ er,
                             float* __restrict__ ebuf, unsigned* __restrict__ menc,
                             int E) {
    const int e = blockIdx.x * blockDim.x + threadIdx.x;
    if (e >= E) return;
    const int d = dst[e];
    float v = el[src[e]] + er[d];
    v = (v > 0.f) ? v : NEG_SLOPE * v;
    ebuf[e] = v;
    atomicMax(&menc[d], enc_f32(v));
}

// ---------- edge pass 2: ee = exp(e - m[dst]); segment sum over dst ----------
__global__ void gat_edge_expsum(const int* __restrict__ dst,
                                float* __restrict__ ebuf,
                                const unsigned* __restrict__ menc,
                                float* __restrict__ s, int E) {
    const int e = blockIdx.x * blockDim.x + threadIdx.x;
    if (e >= E) return;
    const int d = dst[e];
    const float x = __expf(ebuf[e] - dec_f32(menc[d]));
    ebuf[e] = x;            // ebuf now holds ee
    atomicAdd(&s[d], x);
}

// ---------- edge pass 3: out[dst,:] += (ee/s[dst]) * feat[src,:], float4 chunks ----------
__global__ void gat_aggregate(const float* __restrict__ F,
                              const float* __restrict__ ee,
                              const float* __restrict__ s,
                              const int* __restrict__ src, const int* __restrict__ dst,
                              float* __restrict__ out, int E, int D, int FS) {
    const long long gid = (long long)blockIdx.x * blockDim.x + threadIdx.x;
    const int e = (int)(gid >> 4);
    const int p = (int)(gid & 15);
    if (e >= E) return;
    const int c0 = p << 2;
    if (c0 >= D) return;
    const int d = dst[e];
    const float alpha = ee[e] / s[d];
    const float4 v = *(const float4*)(F + (size_t)src[e] * FS + c0);
    float* o = out + (size_t)d * D + c0;
    atomicAdd(o + 0, alpha * v.x);
    atomicAdd(o + 1, alpha * v.y);
    atomicAdd(o + 2, alpha * v.z);
    atomicAdd(o + 3, alpha * v.w);
}

// ---------- h = elu(out + bias), in place ----------
__global__ void gat_bias_elu(float* __restrict__ h, const float* __restrict__ bias,
                             int total, int D) {
    const int i = blockIdx.x * blockDim.x + threadIdx.x;
    if (i >= total) return;
    const float x = h[i] + bias[i & (D - 1)];   // D = 64 (power of two)
    h[i] = (x > 0.f) ? x : (__expf(x) - 1.f);
}

// ---------- out += bias ----------
__global__ void gat_bias_add(float* __restrict__ out, const float* __restrict__ bias,
                             int total, int D) {
    const int i = blockIdx.x * blockDim.x + threadIdx.x;
    if (i >= total) return;
    out[i] += bias[i % D];
}

extern "C" void kernel_launch(void* const* d_in, const int* in_sizes, int n_in,
                              void* d_out, int out_size, void* d_ws, size_t ws_size,
                              hipStream_t stream) {
    const float* X   = (const float*)d_in[0];   // [N,128]
    const int*   src = (const int*)d_in[1];     // [E]
    const int*   dst = (const int*)d_in[2];     // [E]
    const float* W1  = (const float*)d_in[3];   // [128,64]
    const float* al1 = (const float*)d_in[4];   // [64]
    const float* ar1 = (const float*)d_in[5];   // [64]
    const float* b1  = (const float*)d_in[6];   // [64]
    const float* W2  = (const float*)d_in[7];   // [64,40]
    const float* al2 = (const float*)d_in[8];   // [40]
    const float* ar2 = (const float*)d_in[9];   // [40]
    const float* b2  = (const float*)d_in[10];  // [40]

    const int IN = 128, H = 64, CLS = 40, CLSPAD = 48;
    const int N = in_sizes[0] / IN;
    const int E = in_sizes[1];

    // workspace layout (floats)
    float* ws = (float*)d_ws;
    float*    feat1 = ws;                               // N*64
    float*    h     = feat1 + (size_t)N * H;            // N*64 (accum -> elu'd hidden)
    float*    feat2 = h + (size_t)N * H;                // N*48 (padded stride)
    float*    el    = feat2 + (size_t)N * CLSPAD;       // N
    float*    er    = el + N;                           // N
    unsigned* menc  = (unsigned*)(er + N);              // N
    float*    s     = (float*)(menc + N);               // N
    float*    ebuf  = s + N;                            // E (e, then ee)
    float*    Bp1   = ebuf + E;                         // 128*64 = 8192
    float*    Bp2   = Bp1 + (size_t)IN * H;             // 64*48  = 3072

    const int Mtiles = (N + 15) / 16;
    dim3 wblk(32, 8);   // 8 waves per block, one 16-row strip per wave
    const dim3 ggrid((Mtiles + 7) / 8);

    // ================= Layer 1 =================
    {
        hipLaunchKernelGGL(gat_pack_b, dim3((IN / 2 * H + 255) / 256), dim3(256), 0, stream,
                           W1, Bp1, IN, H, H);
        hipLaunchKernelGGL((gat_gemm_wmma<128, 4>), ggrid, wblk, 0, stream,
                           X, Bp1, feat1, N);

        hipLaunchKernelGGL(gat_attn_scores, dim3((N + 255) / 256), dim3(256), 0, stream,
                           feat1, al1, ar1, el, er, N, H, H);

        hipMemsetAsync(menc, 0, (size_t)N * sizeof(unsigned), stream);  // 0 < enc(-inf)
        hipMemsetAsync(s,    0, (size_t)N * sizeof(float), stream);
        hipMemsetAsync(h,    0, (size_t)N * H * sizeof(float), stream);

        hipLaunchKernelGGL(gat_edge_max, dim3((E + 255) / 256), dim3(256), 0, stream,
                           src, dst, el, er, ebuf, menc, E);
        hipLaunchKernelGGL(gat_edge_expsum, dim3((E + 255) / 256), dim3(256), 0, stream,
                           dst, ebuf, menc, s, E);

        const long long work = (long long)E * 16;
        hipLaunchKernelGGL(gat_aggregate, dim3((unsigned)((work + 255) / 256)), dim3(256), 0, stream,
                           feat1, ebuf, s, src, dst, h, E, H, H);

        hipLaunchKernelGGL(gat_bias_elu, dim3((N * H + 255) / 256), dim3(256), 0, stream,
                           h, b1, N * H, H);
    }

    // ================= Layer 2 =================
    {
        float* out = (float*)d_out;   // [N,40] accumulator

        hipLaunchKernelGGL(gat_pack_b, dim3((H / 2 * CLSPAD + 255) / 256), dim3(256), 0, stream,
                           W2, Bp2, H, CLS, CLSPAD);
        hipLaunchKernelGGL((gat_gemm_wmma<64, 3>), ggrid, wblk, 0, stream,
                           h, Bp2, feat2, N);

        hipLaunchKernelGGL(gat_attn_scores, dim3((N + 255) / 256), dim3(256), 0, stream,
                           feat2, al2, ar2, el, er, N, CLS, CLSPAD);

        hipMemsetAsync(menc, 0, (size_t)N * sizeof(unsigned), stream);
        hipMemsetAsync(s,    0, (size_t)N * sizeof(float), stream);
        hipMemsetAsync(out,  0, (size_t)N * CLS * sizeof(float), stream);

        hipLaunchKernelGGL(gat_edge_max, dim3((E + 255) / 256), dim3(256), 0, stream,
                           src, dst, el, er, ebuf, menc, E);
        hipLaunchKernelGGL(gat_edge_expsum, dim3((E + 255) / 256), dim3(256), 0, stream,
                           dst, ebuf, menc, s, E);

        const long long work = (long long)E * 16;   // 10 of 16 float4-parts active
        hipLaunchKernelGGL(gat_aggregate, dim3((unsigned)((work + 255) / 256)), dim3(256), 0, stream,
                           feat2, ebuf, s, src, dst, out, E, CLS, CLSPAD);

        hipLaunchKernelGGL(gat_bias_add, dim3((N * CLS + 255) / 256), dim3(256), 0, stream,
                           out, b2, N * CLS, CLS);
    }
}